// CaptionDecoderRNN_5583457485455
// MI455X (gfx1250) — compile-verified
//
#include <hip/hip_runtime.h>
#include <hip/hip_bf16.h>
#include <math.h>

// ---------------------------------------------------------------------------
// CaptionDecoderRNN for MI455X (gfx1250, wave32, WMMA).
// B=256, T=20, R=100, IN=300, H=1024, V=2048, VOCAB=10000, H4=4096.
// All GEMMs run through a fp32 WMMA (v_wmma_f32_16x16x4_f32) kernel.
// ---------------------------------------------------------------------------

typedef float v2f __attribute__((ext_vector_type(2)));
typedef float v8f __attribute__((ext_vector_type(8)));

#define BB    256
#define TT    20
#define RR    100
#define INDIM 300
#define HID   1024
#define VDIM  2048
#define VOCAB_ 10000
#define H4    4096
#define KV    (VDIM + HID + INDIM + HID)   // 4396 packed A width for LSTM-v
#define KC    (VDIM + HID + HID)           // 4096 packed A width for LSTM-c

// ---------------------------------------------------------------------------
// Generic fp32 WMMA GEMM:  C[M,N] = A[M,K] @ Bstack[K,N] + bias
//   Bstack row k: k < K1 -> B1[k], else B2[k-K1]  (both row-major, ld = N)
//   grid.x = M/64 (exact), grid.y = ceil(N/64), block = 128 (4 waves)
//   Each wave: 16-wide N stripe, 64-tall M supertile (4 wmma accumulators).
//   K, K1, K-K1 must be multiples of 4.
//   Inner loops are branch-free: B1/B2 handled by loop splitting, N-edge
//   handled by clamping the load column (stores/bias remain guarded).
//   Unroll 4 => 16 WMMAs per loop body, pointer bumps amortized into load
//   offsets, ~24 loads in flight for latency hiding at low occupancy.
// ---------------------------------------------------------------------------
__global__ __launch_bounds__(128)
void gemm_wmma_f32(const float* __restrict__ A, int lda,
                   const float* __restrict__ B1, const float* __restrict__ B2,
                   int K1, int K,
                   const float* __restrict__ bias,
                   float* __restrict__ C, int ldc, int N)
{
    const int wave = threadIdx.x >> 5;
    const int lane = threadIdx.x & 31;
    const int half = lane >> 4;       // 0: K {0,1}   1: K {2,3}
    const int l    = lane & 15;
    const int m0   = blockIdx.x * 64;
    const int n0   = blockIdx.y * 64 + wave * 16;
    const int col  = n0 + l;
    const bool colok = (col < N);
    const int col_ld = colok ? col : (N - 1);   // clamped: loads stay in-bounds

    v8f acc0 = {}, acc1 = {}, acc2 = {}, acc3 = {};

    // Per-lane fragment pointers with lane-half K offset folded in.
    const float* a0p = A + (size_t)(m0 +  0 + l) * lda + half * 2;
    const float* a1p = A + (size_t)(m0 + 16 + l) * lda + half * 2;
    const float* a2p = A + (size_t)(m0 + 32 + l) * lda + half * 2;
    const float* a3p = A + (size_t)(m0 + 48 + l) * lda + half * 2;
    const float* b1p = B1 + (size_t)(half * 2) * N + col_ld;
    const float* b2p = B2 + (size_t)(half * 2) * N + col_ld;

#define GEMM_STEP(BP)                                                          \
    {                                                                          \
        v2f bf;                                                                \
        bf[0] = (BP)[0];                                                       \
        bf[1] = (BP)[N];                                                       \
        v2f a0 = *(const v2f*)a0p;                                             \
        v2f a1 = *(const v2f*)a1p;                                             \
        v2f a2 = *(const v2f*)a2p;                                             \
        v2f a3 = *(const v2f*)a3p;                                             \
        acc0 = __builtin_amdgcn_wmma_f32_16x16x4_f32(false, a0, false, bf, (short)0, acc0, false, false); \
        acc1 = __builtin_amdgcn_wmma_f32_16x16x4_f32(false, a1, false, bf, (short)0, acc1, false, false); \
        acc2 = __builtin_amdgcn_wmma_f32_16x16x4_f32(false, a2, false, bf, (short)0, acc2, false, false); \
        acc3 = __builtin_amdgcn_wmma_f32_16x16x4_f32(false, a3, false, bf, (short)0, acc3, false, false); \
        a0p += 4; a1p += 4; a2p += 4; a3p += 4;                                \
        (BP) += (size_t)4 * N;                                                 \
    }

#pragma unroll 4
    for (int k = 0; k < K1; k += 4) GEMM_STEP(b1p);
#pragma unroll 4
    for (int k = K1; k < K; k += 4) GEMM_STEP(b2p);
#undef GEMM_STEP

    if (colok) {
        const float bv = bias ? bias[col] : 0.0f;
#pragma unroll
        for (int j = 0; j < 8; ++j) {
            const int rb = m0 + j + half * 8;
            C[(size_t)(rb +  0) * ldc + col] = acc0[j] + bv;
            C[(size_t)(rb + 16) * ldc + col] = acc1[j] + bv;
            C[(size_t)(rb + 32) * ldc + col] = acc2[j] + bv;
            C[(size_t)(rb + 48) * ldc + col] = acc3[j] + bv;
        }
    }
}

// ---------------------------------------------------------------------------
// feat_mean[b,d] = mean_r features[b,r,d]
// ---------------------------------------------------------------------------
__global__ void feat_mean_kernel(const float* __restrict__ features,
                                 float* __restrict__ fmean)
{
    int idx = blockIdx.x * blockDim.x + threadIdx.x;
    if (idx >= BB * VDIM) return;
    int b = idx >> 11, d = idx & (VDIM - 1);
    const float* f = features + (size_t)b * RR * VDIM + d;
    float s = 0.0f;
    for (int r = 0; r < RR; ++r) s += f[(size_t)r * VDIM];
    fmean[idx] = s * (1.0f / (float)RR);
}

// ---------------------------------------------------------------------------
// x[b, t+1, :] = emb[caption[b,t], :]
// ---------------------------------------------------------------------------
__global__ void embed_kernel(const int* __restrict__ caption,
                             const float* __restrict__ emb,
                             float* __restrict__ x)
{
    int idx = blockIdx.x * blockDim.x + threadIdx.x;
    if (idx >= BB * (TT - 1) * INDIM) return;
    int d  = idx % INDIM;
    int bt = idx / INDIM;
    int b  = bt / (TT - 1);
    int t  = bt % (TT - 1);
    int w  = caption[b * (TT - 1) + t];
    x[(size_t)b * (TT * INDIM) + (size_t)(t + 1) * INDIM + d] = emb[(size_t)w * INDIM + d];
}

// ---------------------------------------------------------------------------
// inp_v[b,:] = [feat_mean(2048) | h_c(1024) | x_t(300) | h_v(1024)]   (4396)
// ---------------------------------------------------------------------------
__global__ void pack_inpv_kernel(const float* __restrict__ fmean,
                                 const float* __restrict__ h_c,
                                 const float* __restrict__ x_t,   // row stride TT*INDIM
                                 const float* __restrict__ h_v,
                                 float* __restrict__ inp_v)
{
    int idx = blockIdx.x * blockDim.x + threadIdx.x;
    if (idx >= BB * KV) return;
    int b = idx / KV, i = idx % KV;
    float v;
    if      (i < VDIM)               v = fmean[b * VDIM + i];
    else if (i < VDIM + HID)         v = h_c[b * HID + (i - VDIM)];
    else if (i < VDIM + HID + INDIM) v = x_t[(size_t)b * (TT * INDIM) + (i - VDIM - HID)];
    else                             v = h_v[b * HID + (i - VDIM - HID - INDIM)];
    inp_v[idx] = v;
}

// ---------------------------------------------------------------------------
// LSTM pointwise: gates[b,4096] (i,f,g,o) -> update c,h in place.
// Optionally streams h into out_h[b*out_stride + j].
// ---------------------------------------------------------------------------
__global__ void lstm_pointwise_kernel(const float* __restrict__ gates,
                                      float* __restrict__ c,
                                      float* __restrict__ h,
                                      float* __restrict__ out_h, int out_stride)
{
    int idx = blockIdx.x * blockDim.x + threadIdx.x;
    if (idx >= BB * HID) return;
    int b = idx >> 10, j = idx & (HID - 1);
    const float* g = gates + (size_t)b * H4;
    float gi = g[j], gf = g[HID + j], gg = g[2 * HID + j], go = g[3 * HID + j];
    float si = 1.0f / (1.0f + expf(-gi));
    float sf = 1.0f / (1.0f + expf(-gf));
    float so = 1.0f / (1.0f + expf(-go));
    float c2 = sf * c[idx] + si * tanhf(gg);
    float h2 = so * tanhf(c2);
    c[idx] = c2;
    h[idx] = h2;
    if (out_h) out_h[(size_t)b * out_stride + j] = h2;
}

// ---------------------------------------------------------------------------
// logit[b,r] = sum_d tanh(va[b,r,d] + ha[b,d]) * W_a[d] + b_a
// one wave per (b,r)
// ---------------------------------------------------------------------------
__global__ void att_logit_kernel(const float* __restrict__ va,
                                 const float* __restrict__ ha,
                                 const float* __restrict__ W_a,
                                 const float* __restrict__ b_a,
                                 float* __restrict__ logit)
{
    int gwave = (blockIdx.x * blockDim.x + threadIdx.x) >> 5;
    int lane  = threadIdx.x & 31;
    if (gwave >= BB * RR) return;
    int b = gwave / RR, r = gwave % RR;
    const float* vr = va + ((size_t)b * RR + r) * HID;
    const float* hr = ha + (size_t)b * HID;
    float s = 0.0f;
    for (int d = lane; d < HID; d += 32)
        s += tanhf(vr[d] + hr[d]) * W_a[d];
    for (int off = 16; off > 0; off >>= 1)
        s += __shfl_down(s, off, 32);
    if (lane == 0) logit[gwave] = s + b_a[0];
}

// ---------------------------------------------------------------------------
// Per batch-row b: softmax over R, store attention to d_out, compute
// feat_hat = (1/R) * sum_r att[r]*features[b,r,:] and pack
// inp_c[b,:] = [feat_hat(2048) | h_v(1024) | h_c(1024)].
// ---------------------------------------------------------------------------
__global__ void att_softmax_feat_kernel(const float* __restrict__ logit,
                                        const float* __restrict__ features,
                                        const float* __restrict__ h_v,
                                        const float* __restrict__ h_c,
                                        float* __restrict__ inp_c,
                                        float* __restrict__ att_out, int t)
{
    int b = blockIdx.x, tid = threadIdx.x;
    __shared__ float sl[RR], sa[RR];
    __shared__ float smax, ssum;

    if (tid < RR) sl[tid] = logit[b * RR + tid];
    __syncthreads();
    if (tid == 0) {
        float m = sl[0];
        for (int r = 1; r < RR; ++r) m = fmaxf(m, sl[r]);
        smax = m;
    }
    __syncthreads();
    if (tid < RR) sa[tid] = expf(sl[tid] - smax);
    __syncthreads();
    if (tid == 0) {
        float s = 0.0f;
        for (int r = 0; r < RR; ++r) s += sa[r];
        ssum = s;
    }
    __syncthreads();
    if (tid < RR) {
        float a = sa[tid] / ssum;
        sa[tid] = a;
        att_out[((size_t)b * TT + t) * RR + tid] = a;
    }
    __syncthreads();

    const float* f  = features + (size_t)b * RR * VDIM;
    float* row = inp_c + (size_t)b * KC;
    for (int d = tid; d < VDIM; d += blockDim.x) {
        float s = 0.0f;
        for (int r = 0; r < RR; ++r) s += f[(size_t)r * VDIM + d] * sa[r];
        row[d] = s * (1.0f / (float)RR);
    }
    for (int j = tid; j < HID; j += blockDim.x) {
        row[VDIM + j]       = h_v[b * HID + j];
        row[VDIM + HID + j] = h_c[b * HID + j];
    }
}

// ---------------------------------------------------------------------------
// Workspace layout (floats).  Total ~38.1M floats (~152 MB).
// ---------------------------------------------------------------------------
static constexpr size_t OFF_X      = 0;
static constexpr size_t SZ_X       = (size_t)BB * TT * INDIM;        // 1,536,000
static constexpr size_t OFF_FMEAN  = OFF_X + SZ_X;
static constexpr size_t SZ_FMEAN   = (size_t)BB * VDIM;
static constexpr size_t OFF_VA     = OFF_FMEAN + SZ_FMEAN;
static constexpr size_t SZ_VA      = (size_t)BB * RR * HID;          // 26,214,400
static constexpr size_t OFF_INPV   = OFF_VA + SZ_VA;
static constexpr size_t SZ_INPV    = (size_t)BB * KV;
static constexpr size_t OFF_INPC   = OFF_INPV + SZ_INPV;
static constexpr size_t SZ_INPC    = (size_t)BB * KC;
static constexpr size_t OFF_GATES  = OFF_INPC + SZ_INPC;
static constexpr size_t SZ_GATES   = (size_t)BB * H4;
static constexpr size_t OFF_STATES = OFF_GATES + SZ_GATES;           // h_v,c_v,h_c,c_c
static constexpr size_t SZ_STATE   = (size_t)BB * HID;
static constexpr size_t OFF_HA     = OFF_STATES + 4 * SZ_STATE;
static constexpr size_t OFF_LOGIT  = OFF_HA + SZ_STATE;
static constexpr size_t OFF_OUTS   = OFF_LOGIT + (size_t)BB * RR;
static constexpr size_t SZ_OUTS    = (size_t)BB * TT * HID;

extern "C" void kernel_launch(void* const* d_in, const int* in_sizes, int n_in,
                              void* d_out, int out_size, void* d_ws, size_t ws_size,
                              hipStream_t stream)
{
    const int*   caption  = (const int*)  d_in[0];
    const float* features = (const float*)d_in[1];
    const float* emb      = (const float*)d_in[2];
    const float* W_vin    = (const float*)d_in[3];
    const float* b_vin    = (const float*)d_in[4];
    const float* W_ih_v   = (const float*)d_in[5];
    const float* W_hh_v   = (const float*)d_in[6];
    const float* b_v      = (const float*)d_in[7];
    const float* W_va     = (const float*)d_in[8];
    const float* b_va     = (const float*)d_in[9];
    const float* W_ha     = (const float*)d_in[10];
    const float* b_ha     = (const float*)d_in[11];
    const float* W_a      = (const float*)d_in[12];
    const float* b_a      = (const float*)d_in[13];
    const float* W_ih_c   = (const float*)d_in[14];
    const float* W_hh_c   = (const float*)d_in[15];
    const float* b_c      = (const float*)d_in[16];
    const float* W_cls    = (const float*)d_in[17];
    const float* b_cls    = (const float*)d_in[18];

    float* ws     = (float*)d_ws;
    float* x      = ws + OFF_X;
    float* fmean  = ws + OFF_FMEAN;
    float* va     = ws + OFF_VA;
    float* inp_v  = ws + OFF_INPV;
    float* inp_c  = ws + OFF_INPC;
    float* gates  = ws + OFF_GATES;
    float* h_v    = ws + OFF_STATES + 0 * SZ_STATE;
    float* c_v    = ws + OFF_STATES + 1 * SZ_STATE;
    float* h_c    = ws + OFF_STATES + 2 * SZ_STATE;
    float* c_c    = ws + OFF_STATES + 3 * SZ_STATE;
    float* ha     = ws + OFF_HA;
    float* logit  = ws + OFF_LOGIT;
    float* outs   = ws + OFF_OUTS;

    float* repr    = (float*)d_out;
    float* att_out = repr + (size_t)BB * TT * VOCAB_;

    // zero LSTM state (ws is not re-poisoned between replays, so do it every call)
    hipMemsetAsync(ws + OFF_STATES, 0, 4 * SZ_STATE * sizeof(float), stream);

    // feat_mean, embedding gather
    feat_mean_kernel<<<(BB * VDIM + 255) / 256, 256, 0, stream>>>(features, fmean);
    embed_kernel<<<(BB * (TT - 1) * INDIM + 255) / 256, 256, 0, stream>>>(caption, emb, x);

    // x[:,0,:] = feat_mean @ W_vin + b_vin   (M=256, K=2048, N=300)
    {
        dim3 g(BB / 64, (INDIM + 63) / 64);
        gemm_wmma_f32<<<g, 128, 0, stream>>>(fmean, VDIM, W_vin, W_vin, VDIM, VDIM,
                                             b_vin, x, TT * INDIM, INDIM);
    }
    // va = features @ W_va + b_va            (M=25600, K=2048, N=1024)
    {
        dim3 g((BB * RR) / 64, HID / 64);
        gemm_wmma_f32<<<g, 128, 0, stream>>>(features, VDIM, W_va, W_va, VDIM, VDIM,
                                             b_va, va, HID, HID);
    }

    for (int t = 0; t < TT; ++t) {
        // pack [feat_mean | h_c | x_t | h_v]  and fused dual-weight GEMM (+W_hh_v)
        pack_inpv_kernel<<<(BB * KV + 255) / 256, 256, 0, stream>>>(
            fmean, h_c, x + (size_t)t * INDIM, h_v, inp_v);
        {
            dim3 g(BB / 64, H4 / 64);
            gemm_wmma_f32<<<g, 128, 0, stream>>>(inp_v, KV, W_ih_v, W_hh_v,
                                                 VDIM + HID + INDIM, KV,
                                                 b_v, gates, H4, H4);
        }
        lstm_pointwise_kernel<<<(BB * HID) / 256, 256, 0, stream>>>(gates, c_v, h_v, nullptr, 0);

        // ha = h_v @ W_ha + b_ha             (M=256, K=1024, N=1024)
        {
            dim3 g(BB / 64, HID / 64);
            gemm_wmma_f32<<<g, 128, 0, stream>>>(h_v, HID, W_ha, W_ha, HID, HID,
                                                 b_ha, ha, HID, HID);
        }

        att_logit_kernel<<<(BB * RR * 32 + 255) / 256, 256, 0, stream>>>(va, ha, W_a, b_a, logit);
        att_softmax_feat_kernel<<<BB, 128, 0, stream>>>(logit, features, h_v, h_c,
                                                        inp_c, att_out, t);

        // gates = [feat_hat|h_v] @ W_ih_c + h_c @ W_hh_c + b_c
        {
            dim3 g(BB / 64, H4 / 64);
            gemm_wmma_f32<<<g, 128, 0, stream>>>(inp_c, KC, W_ih_c, W_hh_c,
                                                 VDIM + HID, KC,
                                                 b_c, gates, H4, H4);
        }
        lstm_pointwise_kernel<<<(BB * HID) / 256, 256, 0, stream>>>(
            gates, c_c, h_c, outs + (size_t)t * HID, TT * HID);
    }

    // repr = outputs @ W_cls + b_cls         (M=5120, K=1024, N=10000)
    {
        dim3 g((BB * TT) / 64, (VOCAB_ + 63) / 64);
        gemm_wmma_f32<<<g, 128, 0, stream>>>(outs, HID, W_cls, W_cls, HID, HID,
                                             b_cls, repr, VOCAB_, VOCAB_);
    }
}